// Model_62955630625110
// MI455X (gfx1250) — compile-verified
//
#include <hip/hip_runtime.h>

typedef float v2f __attribute__((ext_vector_type(2)));
typedef float v8f __attribute__((ext_vector_type(8)));

#define WAVES_PER_BLOCK 8
#define RAYS_PER_WAVE  16
#define RAYS_PER_BLOCK (WAVES_PER_BLOCK * RAYS_PER_WAVE)

// One wave handles a 16-ray x 256-sphere tile strip.
// b and c (N,M) coefficient tiles are produced by V_WMMA_F32_16X16X4_F32.
__global__ __launch_bounds__(256) void raytrace_wmma(
    const float* __restrict__ RO,  // (N,3) ray origins
    const float* __restrict__ RD,  // (N,3) ray directions
    const float* __restrict__ SC,  // (M,3) sphere centers
    const float* __restrict__ SR,  // (M,)  sphere radii
    float* __restrict__ out,       // [t_hit N | idx N | hp 3N | nrm 3N]
    int nRays, int nSph)
{
  __shared__ float4 sph[256];      // (-2cx, -2cy, -2cz, |c|^2 - r^2)

  const int tid = threadIdx.x;
  if (tid < nSph) {
    float cx = SC[tid*3+0], cy = SC[tid*3+1], cz = SC[tid*3+2];
    float rr = SR[tid];
    float4 s;
    s.x = -2.0f*cx; s.y = -2.0f*cy; s.z = -2.0f*cz;
    s.w = cx*cx + cy*cy + cz*cz - rr*rr;
    sph[tid] = s;
  }
  __syncthreads();

  const int wave = tid >> 5;
  const int lane = tid & 31;
  const int rayBase = (blockIdx.x * WAVES_PER_BLOCK + wave) * RAYS_PER_WAVE;
  if (rayBase >= nRays) return;           // wave-uniform

  const int m  = lane & 15;               // ray row this lane feeds (A-matrix)
  const int hi = lane >> 4;               // lane half selects K pair {0,1} vs {2,3}
  const int ray = rayBase + m;

  const float ox = RO[ray*3+0], oy = RO[ray*3+1], oz = RO[ray*3+2];
  const float dx = RD[ray*3+0], dy = RD[ray*3+1], dz = RD[ray*3+2];
  const float ddo = dx*ox + dy*oy + dz*oz;     // d.o
  const float on2 = ox*ox + oy*oy + oz*oz;     // |o|^2
  const float av  = dx*dx + dy*dy + dz*dz;     // a = d.d
  const float fourA = 4.0f * av;
  const float inv2a = __builtin_amdgcn_rcpf(2.0f * av);   // v_rcp_f32

  // A-matrix (16x4 f32): VGPR0 = K(0|2), VGPR1 = K(1|3) per lane half.
  v2f Ab; Ab.x = hi ? dz : dx;  Ab.y = hi ? ddo  : dy;   // [dx,dy,dz,d.o]
  v2f Ac; Ac.x = hi ? oz : ox;  Ac.y = hi ? 1.0f : oy;   // [ox,oy,oz,1]

  // Per-ray constants rearranged into C/D layout: VGPR r <-> M = r + 8*hi.
  float cacc[8], fA[8], i2a[8];
  #pragma unroll
  for (int r = 0; r < 8; ++r) {
    int src = r + (hi << 3);
    cacc[r] = __shfl(on2,   src, 32);
    fA[r]   = __shfl(fourA, src, 32);
    i2a[r]  = __shfl(inv2a, src, 32);
  }

  v8f Cc;
  #pragma unroll
  for (int r = 0; r < 8; ++r) Cc[r] = cacc[r];

  float tmin[8]; int imin[8];
  #pragma unroll
  for (int r = 0; r < 8; ++r) { tmin[r] = __builtin_inff(); imin[r] = 0x7fffffff; }

  const int n0 = lane & 15;

  if (nSph == 256) {
    // ---- Specialized path: 16 tiles, fully register-resident B operands ----
    // Per lane per tile only 3 floats survive the half-select
    // (Bb.x == Bc.x). 48 VGPRs total; all 16 ds_load_b128 issue
    // back-to-back so LDS latency is pipelined once, up front.
    float B0[16], B1[16], C1[16];
    #pragma unroll
    for (int t = 0; t < 16; ++t) {
      float4 s = sph[n0 + (t << 4)];
      B0[t] = hi ? s.z : s.x;     // [-2c] component pair (K0|K2)
      B1[t] = hi ? 2.0f : s.y;    // b-matrix K1|K3: [-2cy, 2]
      C1[t] = hi ? s.w  : s.y;    // c-matrix K1|K3: [-2cy, |c|^2-r^2]
    }

    #pragma unroll
    for (int t = 0; t < 16; ++t) {
      const int n = n0 + (t << 4);
      v2f Bb; Bb.x = B0[t]; Bb.y = B1[t];
      v2f Bc; Bc.x = B0[t]; Bc.y = C1[t];
      v8f Cz = {};
      // b = 2 d.o - 2 d.c ; c = |o|^2 - 2 o.c + (|c|^2 - r^2)
      v8f Db = __builtin_amdgcn_wmma_f32_16x16x4_f32(
          false, Ab, false, Bb, (short)0, Cz, false, false);
      v8f Dc = __builtin_amdgcn_wmma_f32_16x16x4_f32(
          false, Ac, false, Bc, (short)0, Cc, false, false);

      #pragma unroll
      for (int r = 0; r < 8; ++r) {
        float b = Db[r], c = Dc[r];
        float disc = fmaf(b, b, -(fA[r] * c));      // b^2 - 4ac
        // disc<0 -> sqrt NaN; disc>=0 predicate forces tm=inf anyway.
        float sd   = __builtin_amdgcn_sqrtf(disc);  // raw v_sqrt_f32
        float t1   = (-b - sd) * i2a[r];
        float t2   = (-b + sd) * i2a[r];
        float tt   = (t1 > 0.0f) ? t1 : t2;
        float tm   = ((disc >= 0.0f) && (tt > 0.0f)) ? tt : __builtin_inff();
        if (tm < tmin[r]) { tmin[r] = tm; imin[r] = n; }
      }
    }
  } else {
    // ---- Generic fallback for other sphere counts ----
    const int ntiles = nSph >> 4;
    for (int t = 0; t < ntiles; ++t) {
      const int n = n0 + (t << 4);
      const float4 s = sph[n];
      v2f Bb; Bb.x = hi ? s.z : s.x;  Bb.y = hi ? 2.0f : s.y;
      v2f Bc; Bc.x = hi ? s.z : s.x;  Bc.y = hi ? s.w  : s.y;
      v8f Cz = {};
      v8f Db = __builtin_amdgcn_wmma_f32_16x16x4_f32(
          false, Ab, false, Bb, (short)0, Cz, false, false);
      v8f Dc = __builtin_amdgcn_wmma_f32_16x16x4_f32(
          false, Ac, false, Bc, (short)0, Cc, false, false);
      #pragma unroll
      for (int r = 0; r < 8; ++r) {
        float b = Db[r], c = Dc[r];
        float disc = fmaf(b, b, -(fA[r] * c));
        float sd   = __builtin_amdgcn_sqrtf(disc);
        float t1   = (-b - sd) * i2a[r];
        float t2   = (-b + sd) * i2a[r];
        float tt   = (t1 > 0.0f) ? t1 : t2;
        float tm   = ((disc >= 0.0f) && (tt > 0.0f)) ? tt : __builtin_inff();
        if (tm < tmin[r]) { tmin[r] = tm; imin[r] = n; }
      }
    }
  }

  // min/argmin across the 16 lanes of each half (xor 1,2,4,8 stays in-half).
  #pragma unroll
  for (int r = 0; r < 8; ++r) {
    float tv = tmin[r]; int iv = imin[r];
    #pragma unroll
    for (int off = 1; off < 16; off <<= 1) {
      float to = __shfl_xor(tv, off, 32);
      int   io = __shfl_xor(iv, off, 32);
      if ((to < tv) || (to == tv && io < iv)) { tv = to; iv = io; }
    }
    tmin[r] = tv; imin[r] = iv;
  }

  // Writer lanes: own loaded ray m equals the reduced ray M = (lane&7) + 8*hi.
  if (hi == (m >> 3)) {
    const int r  = lane & 7;
    const float tv = tmin[r];
    const int   iv = imin[r];
    const bool hit = (tv < 3.0e38f);

    out[ray] = tv;
    reinterpret_cast<int*>(out + nRays)[ray] = hit ? iv : -1;

    const float hx = fmaf(tv, dx, ox);
    const float hy = fmaf(tv, dy, oy);
    const float hz = fmaf(tv, dz, oz);
    float* hp = out + (size_t)2 * nRays;
    hp[ray*3+0] = hx; hp[ray*3+1] = hy; hp[ray*3+2] = hz;

    float nx = 0.0f, ny = 0.0f, nz = 0.0f;
    if (hit) {
      float4 sc = sph[iv];
      nx = hx + 0.5f*sc.x;     // hx - cx  (sc.x = -2cx)
      ny = hy + 0.5f*sc.y;
      nz = hz + 0.5f*sc.z;
      float inv = __builtin_amdgcn_rsqf(nx*nx + ny*ny + nz*nz);  // v_rsq_f32
      nx *= inv; ny *= inv; nz *= inv;
    }
    float* nm = out + (size_t)5 * nRays;
    nm[ray*3+0] = nx; nm[ray*3+1] = ny; nm[ray*3+2] = nz;
  }
}

extern "C" void kernel_launch(void* const* d_in, const int* in_sizes, int n_in,
                              void* d_out, int out_size, void* d_ws, size_t ws_size,
                              hipStream_t stream) {
  const float* RO = (const float*)d_in[0];
  const float* RD = (const float*)d_in[1];
  const float* SC = (const float*)d_in[2];
  const float* SR = (const float*)d_in[3];
  float* out = (float*)d_out;
  const int nRays = in_sizes[0] / 3;
  const int nSph  = in_sizes[3];
  const int blocks = (nRays + RAYS_PER_BLOCK - 1) / RAYS_PER_BLOCK;
  raytrace_wmma<<<blocks, 256, 0, stream>>>(RO, RD, SC, SR, out, nRays, nSph);
}